// RRNNforGRUCell_57312043598096
// MI455X (gfx1250) — compile-verified
//
#include <hip/hip_runtime.h>
#include <hip/hip_bf16.h>
#include <math.h>

#define H 2048
#define NCMAX 12
#define NPAIRS 66          // NCMAX choose 2
#define NRES (NPAIRS * 8)  // 528: pairs * 4k * 2bf
#define NENT_MAX (NRES * 4)
#define MULT 0.001f

typedef float v2f __attribute__((ext_vector_type(2)));
typedef float v8f __attribute__((ext_vector_type(8)));

// ---------------- workspace layout (floats, then ints) ----------------
// Cmat   : 16*H
// CLR    : 2*4*16*H   ([mat][k][row][h])
// resbuf : NRES*H
// red    : NRES*4     (wsig, wtanh, wres, maxabs)
// scor   : NENT_MAX
// G      : 9*H
// second : 9*H
// sumw   : 1
// coef   : 9
// state (ints): comp_len[1], comp_src[9], rowsrc[16], valid[528],
//               extra[528], second_sel[2], ent_rid[2112], ent_act[2112]
#define OFF_CMAT   0
#define OFF_CLR    (OFF_CMAT + 16 * H)
#define OFF_RES    (OFF_CLR + 2 * 4 * 16 * H)
#define OFF_RED    (OFF_RES + NRES * H)
#define OFF_SCOR   (OFF_RED + NRES * 4)
#define OFF_G      (OFF_SCOR + NENT_MAX)
#define OFF_SECOND (OFF_G + 9 * H)
#define OFF_SUMW   (OFF_SECOND + 9 * H)
#define OFF_COEF   (OFF_SUMW + 1)
#define OFF_END_F  (OFF_COEF + 9)
#define N_STATE_INT (1 + 9 + 16 + NRES + NRES + 2 + NENT_MAX + NENT_MAX)
#define WS_NEEDED ((size_t)OFF_END_F * 4 + (size_t)N_STATE_INT * 4)

// state sub-offsets (in ints)
#define ST_COMPLEN 0
#define ST_COMPSRC 1
#define ST_ROWSRC  10
#define ST_VALID   26
#define ST_EXTRA   (26 + NRES)
#define ST_SECSEL  (26 + 2 * NRES)
#define ST_ENTRID  (28 + 2 * NRES)
#define ST_ENTACT  (28 + 2 * NRES + NENT_MAX)

__device__ __forceinline__ void pair_decode(int p, int* pi, int* pj) {
    int i = 0, rem = p, span = NCMAX - 1;
    while (rem >= span) { rem -= span; i++; span--; }
    *pi = i;
    *pj = i + 1 + rem;
}

// ---------------- init: state + sum(w) ----------------
__global__ void init_state_kernel(int* state) {
    if (threadIdx.x == 0) {
        state[ST_COMPLEN] = 0;
        int* rowsrc = state + ST_ROWSRC;
        rowsrc[0] = 1;  // x
        rowsrc[1] = 2;  // h_prev
        for (int q = 2; q < 16; q++) rowsrc[q] = 0;  // zero rows
    }
}

__global__ void sumw_kernel(const float* __restrict__ w, float* sumw) {
    __shared__ float sm[256];
    float s = 0.f;
    for (int h = threadIdx.x; h < H; h += 256) s += w[h];
    sm[threadIdx.x] = s;
    __syncthreads();
    for (int o = 128; o > 0; o >>= 1) {
        if (threadIdx.x < o) sm[threadIdx.x] += sm[threadIdx.x + o];
        __syncthreads();
    }
    if (threadIdx.x == 0) *sumw = sm[0];
}

// ---------------- build candidate matrix (16 x H, zero padded) ----------------
__global__ void build_cmat_kernel(float* __restrict__ Cmat,
                                  const float* __restrict__ x,
                                  const float* __restrict__ hprev,
                                  const float* __restrict__ G,
                                  const int* __restrict__ state) {
    int row = blockIdx.y;
    int col = blockIdx.x * 256 + threadIdx.x;
    int code = state[ST_ROWSRC + row];
    float v = 0.f;
    if (code == 1)      v = x[col];
    else if (code == 2) v = hprev[col];
    else if (code >= 3) v = G[(size_t)(code - 3) * H + col];
    Cmat[row * H + col] = v;
}

// ---------------- WMMA GEMM: CLR[mat][k] = Cmat(16xH) @ {L,R}[k](HxH) --------
// grid: (H/16, 4, 2), block: 256 (8 waves, K split, LDS reduce)
__global__ void gemm_clr_kernel(const float* __restrict__ Cmat,
                                const float* __restrict__ L,
                                const float* __restrict__ R,
                                float* __restrict__ CLR) {
    const int ntile = blockIdx.x;        // 0..127
    const int k     = blockIdx.y;        // 0..3
    const int mat   = blockIdx.z;        // 0: L, 1: R
    const float* __restrict__ B = (mat == 0 ? L : R) + (size_t)k * H * H;

    const int n0    = ntile * 16;
    const int wave  = threadIdx.x >> 5;
    const int lane  = threadIdx.x & 31;
    const int m     = lane & 15;
    const int khalf = (lane >> 4) << 1;  // 0 or 2
    const int n     = n0 + (lane & 15);

    v8f acc = {};
    const int kstart = wave * (H / 8);   // 256-wide K slice per wave
    for (int kk = 0; kk < H / 8; kk += 4) {
        const int kb = kstart + kk;
        v2f a, b;
        a.x = Cmat[m * H + kb + khalf];
        a.y = Cmat[m * H + kb + khalf + 1];
        b.x = B[(size_t)(kb + khalf) * H + n];
        b.y = B[(size_t)(kb + khalf + 1) * H + n];
        if (kk + 4 < H / 8)
            __builtin_prefetch(&B[(size_t)(kb + 4 + khalf) * H + n], 0, 0);
        acc = __builtin_amdgcn_wmma_f32_16x16x4_f32(
            false, a, false, b, (short)0, acc, false, false);
    }

    __shared__ float lds[8][32][8];
#pragma unroll
    for (int i = 0; i < 8; i++) lds[wave][lane][i] = acc[i];
    __syncthreads();

    // 256 threads: one (lane2, vi) each, sum over 8 waves
    const int lane2 = threadIdx.x & 31;
    const int vi    = threadIdx.x >> 5;
    float s = 0.f;
#pragma unroll
    for (int wv = 0; wv < 8; wv++) s += lds[wv][lane2][vi];
    const int mrow = vi + ((lane2 >> 4) << 3);
    const int ncol = n0 + (lane2 & 15);
    CLR[(((size_t)mat * 4 + k) * 16 + mrow) * H + ncol] = s;
}

// ---------------- res vectors + per-res reductions ----------------
// grid: (NRES), block 256
__global__ void res_kernel(const float* __restrict__ CLR,
                           const float* __restrict__ bvec,
                           const float* __restrict__ w,
                           float* __restrict__ resbuf,
                           float* __restrict__ red) {
    const int rid = blockIdx.x;
    const int p   = rid >> 3;
    const int k   = (rid >> 1) & 3;
    const int bf  = rid & 1;   // 0=mul, 1=add
    int i, j;
    pair_decode(p, &i, &j);

    const float* __restrict__ CLi = CLR + (((size_t)0 * 4 + k) * 16 + i) * H;
    const float* __restrict__ CRj = CLR + (((size_t)1 * 4 + k) * 16 + j) * H;
    const float* __restrict__ bk  = bvec + k * H;

    float wsig = 0.f, wtan = 0.f, wres = 0.f, mab = 0.f;
    for (int h = threadIdx.x; h < H; h += 256) {
        float rv = bf ? (CLi[h] + CRj[h] + bk[h]) : (CLi[h] * CRj[h] + bk[h]);
        resbuf[(size_t)rid * H + h] = rv;
        float wh = w[h];
        wsig += wh / (1.f + __expf(-rv));
        wtan += wh * tanhf(rv);
        wres += wh * rv;
        mab = fmaxf(mab, fabsf(rv));
    }
    __shared__ float s0[256], s1[256], s2[256], s3[256];
    const int t = threadIdx.x;
    s0[t] = wsig; s1[t] = wtan; s2[t] = wres; s3[t] = mab;
    __syncthreads();
    for (int o = 128; o > 0; o >>= 1) {
        if (t < o) {
            s0[t] += s0[t + o];
            s1[t] += s1[t + o];
            s2[t] += s2[t + o];
            s3[t] = fmaxf(s3[t], s3[t + o]);
        }
        __syncthreads();
    }
    if (t == 0) {
        red[rid * 4 + 0] = s0[0];
        red[rid * 4 + 1] = s1[0];
        red[rid * 4 + 2] = s2[0];
        red[rid * 4 + 3] = s3[0];
    }
}

// ---------------- planner: serial control flow, 1 thread ----------------
__global__ void planner_kernel(const float* __restrict__ red,
                               const float* __restrict__ sumw_p,
                               int r, int* __restrict__ state,
                               float* __restrict__ scor,
                               float* __restrict__ coef_out) {
    if (threadIdx.x != 0) return;
    int* comp_src = state + ST_COMPSRC;
    int* rowsrc   = state + ST_ROWSRC;
    int* valid    = state + ST_VALID;
    int* extra    = state + ST_EXTRA;
    int* secsel   = state + ST_SECSEL;
    int* ent_rid  = state + ST_ENTRID;
    int* ent_act  = state + ST_ENTACT;

    int lc  = state[ST_COMPLEN];
    int nc  = 3 + lc;
    int nri = 9 - r;
    float sumw = *sumw_p;

    for (int q = 0; q < NRES; q++) { valid[q] = 0; extra[q] = 0; }

    int nE = 0;
    for (int i = 0; i < nc - 1; i++) {
        for (int j = i + 1; j < nc; j++) {
            int flag;
            if (nri == lc - 1)      flag = (i >= 3);
            else if (nri == lc)     flag = (j >= 3);
            else                    flag = (nri > lc);
            if (!flag) continue;
            int p = i * (2 * NCMAX - i - 1) / 2 + (j - i - 1);
            for (int k = 0; k < 4; k++) {
                for (int bf = 0; bf < 2; bf++) {
                    int rid = p * 8 + k * 2 + bf;
                    valid[rid] = 1;
                    float wsig = red[rid * 4 + 0];
                    float wtan = red[rid * 4 + 1];
                    float wres = red[rid * 4 + 2];
                    float mab  = red[rid * 4 + 3];
                    int hasExtra = (mab < 1.0f);
                    extra[rid] = hasExtra;
                    scor[nE] = MULT * wsig; ent_rid[nE] = rid; ent_act[nE] = 0; nE++;
                    scor[nE] = MULT * wtan; ent_rid[nE] = rid; ent_act[nE] = 1; nE++;
                    if (hasExtra) {
                        scor[nE] = sumw - wres; ent_rid[nE] = rid; ent_act[nE] = 2; nE++;
                        scor[nE] = wres;        ent_rid[nE] = rid; ent_act[nE] = 3; nE++;
                    }
                }
            }
        }
    }

    // argmax (first occurrence)
    int max_idx = 0; float mx = scor[0];
    for (int e = 1; e < nE; e++) if (scor[e] > mx) { mx = scor[e]; max_idx = e; }

    // "second" index = rank of entry #1 in ascending sort (replicates ref bug)
    float s1 = scor[1];
    int rank = 0;
    for (int e = 0; e < nE; e++)
        if (scor[e] < s1) rank++;
    int second_idx = rank;

    // softmax prob of entry index (nc-2)
    double denom = 0.0;
    for (int e = 0; e < nE; e++) denom += exp((double)(scor[e] - mx));
    coef_out[0] = (float)(exp((double)(scor[nc - 2] - mx)) / denom);

    secsel[0] = ent_rid[second_idx];
    secsel[1] = ent_act[second_idx];

    // comp_src list surgery based on winning pair
    int ii, jj;
    pair_decode(ent_rid[max_idx] >> 3, &ii, &jj);
    if (ii > 2 && jj > 2) {
        comp_src[ii - 3] = r;
        for (int q = jj - 3; q < lc - 1; q++) comp_src[q] = comp_src[q + 1];
        lc--;
    } else if (ii <= 2 && jj > 2) {
        comp_src[jj - 3] = r;
    } else {
        comp_src[lc++] = r;
    }
    state[ST_COMPLEN] = lc;

    // rowsrc for next iteration's candidate matrix
    rowsrc[0] = 1; rowsrc[1] = 2; rowsrc[2] = 0;
    for (int q = 0; q < 13; q++)
        rowsrc[3 + q] = (q < lc) ? (3 + comp_src[q]) : 0;
}

// ---------------- combine: G[r] and second[r] ----------------
// grid H/256, block 256
__global__ void combine_kernel(const float* __restrict__ resbuf,
                               const int* __restrict__ state,
                               const float* __restrict__ coef,
                               float* __restrict__ G,
                               float* __restrict__ second, int r) {
    const int h = blockIdx.x * 256 + threadIdx.x;
    const int* valid = state + ST_VALID;
    const int* extra = state + ST_EXTRA;
    const int rid2 = state[ST_SECSEL + 0];
    const int act2 = state[ST_SECSEL + 1];

    float sum = 0.f;
    for (int rid = 0; rid < NRES; rid++) {
        if (!valid[rid]) continue;
        float rv = resbuf[(size_t)rid * H + h];
        // sigmoid + tanh always; minus + identity sum to exactly 1.0
        sum += MULT * (1.f / (1.f + __expf(-rv)) + tanhf(rv));
        if (extra[rid]) sum += 1.0f;
    }
    G[(size_t)r * H + h] = coef[r] * sum;

    float rv2 = resbuf[(size_t)rid2 * H + h];
    float sv;
    if (act2 == 0)      sv = MULT / (1.f + __expf(-rv2));
    else if (act2 == 1) sv = MULT * tanhf(rv2);
    else if (act2 == 2) sv = 1.f - rv2;
    else                sv = rv2;
    second[(size_t)r * H + h] = sv;
}

// ---------------- finalize ----------------
__global__ void copy_out_kernel(const float* __restrict__ G,
                                float* __restrict__ out) {
    const int t = blockIdx.x * 256 + threadIdx.x;  // 0 .. 20479
    if (t < H) out[t] = G[8 * H + t];              // h_next = G[8]
    else       out[t] = G[t - H];                  // G stack (9*H)
}

__global__ void dots_kernel(const float* __restrict__ G,
                            const float* __restrict__ second,
                            const float* __restrict__ w,
                            float* __restrict__ out) {
    const int row = blockIdx.x;  // 0..17
    const float* src = (row < 9) ? (G + (size_t)row * H)
                                 : (second + (size_t)(row - 9) * H);
    __shared__ float sm[256];
    float s = 0.f;
    for (int h = threadIdx.x; h < H; h += 256) s += src[h] * w[h];
    sm[threadIdx.x] = s;
    __syncthreads();
    for (int o = 128; o > 0; o >>= 1) {
        if (threadIdx.x < o) sm[threadIdx.x] += sm[threadIdx.x + o];
        __syncthreads();
    }
    if (threadIdx.x == 0) out[10 * H + row] = sm[0];
}

extern "C" void kernel_launch(void* const* d_in, const int* in_sizes, int n_in,
                              void* d_out, int out_size, void* d_ws, size_t ws_size,
                              hipStream_t stream) {
    if (ws_size < WS_NEEDED) return;

    const float* x  = (const float*)d_in[0];
    const float* hp = (const float*)d_in[1];
    const float* L  = (const float*)d_in[2];
    const float* R  = (const float*)d_in[3];
    const float* b  = (const float*)d_in[4];
    const float* w  = (const float*)d_in[5];
    float* out = (float*)d_out;

    float* wsf    = (float*)d_ws;
    float* Cmat   = wsf + OFF_CMAT;
    float* CLR    = wsf + OFF_CLR;
    float* resbuf = wsf + OFF_RES;
    float* red    = wsf + OFF_RED;
    float* scor   = wsf + OFF_SCOR;
    float* G      = wsf + OFF_G;
    float* second = wsf + OFF_SECOND;
    float* sumw   = wsf + OFF_SUMW;
    float* coef   = wsf + OFF_COEF;
    int*   state  = (int*)(wsf + OFF_END_F);

    init_state_kernel<<<1, 32, 0, stream>>>(state);
    sumw_kernel<<<1, 256, 0, stream>>>(w, sumw);

    for (int r = 0; r < 9; r++) {
        build_cmat_kernel<<<dim3(H / 256, 16), 256, 0, stream>>>(Cmat, x, hp, G, state);
        gemm_clr_kernel<<<dim3(H / 16, 4, 2), 256, 0, stream>>>(Cmat, L, R, CLR);
        res_kernel<<<NRES, 256, 0, stream>>>(CLR, b, w, resbuf, red);
        planner_kernel<<<1, 32, 0, stream>>>(red, sumw, r, state, scor, coef + r);
        combine_kernel<<<H / 256, 256, 0, stream>>>(resbuf, state, coef, G, second, r);
    }

    copy_out_kernel<<<(10 * H) / 256, 256, 0, stream>>>(G, out);
    dots_kernel<<<18, 256, 0, stream>>>(G, second, w, out);
}